// UC_MIL_24043226923616
// MI455X (gfx1250) — compile-verified
//
#include <hip/hip_runtime.h>

typedef float v2f __attribute__((ext_vector_type(2)));
typedef float v8f __attribute__((ext_vector_type(8)));

constexpr int Bb   = 8;
constexpr int Nn   = 4096;
constexpr int Dd   = 768;
constexpr int TOPK = 50;

constexpr int BT   = 64;       // block tile: 64 n-rows x 64 m-cols
constexpr int KC   = 32;       // K chunk staged in LDS
constexpr int LSTR = KC + 4;   // padded LDS row stride (floats) -> conflict-free
constexpr int NC   = Dd / KC;  // 24 chunks (even)

// ---------------- Stage 1: feature means mu[b,d] = mean_n X[b,n,d] ----------------
__global__ void mean_kernel(const float* __restrict__ X, float* __restrict__ mu) {
  int i = blockIdx.x * blockDim.x + threadIdx.x;
  if (i >= Bb * Dd) return;
  int b = i / Dd, d = i - b * Dd;
  const float* p = X + (size_t)b * Nn * Dd + d;
  float s = 0.f;
  for (int n = 0; n < Nn; ++n) s += p[(size_t)n * Dd];
  mu[i] = s * (1.0f / Nn);
}

// ---------------- Stage 2: per-row var, X*W, w1 = 1/var (one wave per row) --------
__global__ __launch_bounds__(256)
void row_stats_kernel(const float* __restrict__ X, const float* __restrict__ W,
                      const float* __restrict__ mu, float* __restrict__ var,
                      float* __restrict__ xw, float* __restrict__ w1) {
  int row  = blockIdx.x * 8 + (threadIdx.x >> 5);   // global row in [0, B*N)
  int lane = threadIdx.x & 31;
  int b = row / Nn;
  const float* xp = X + (size_t)row * Dd;
  const float* mp = mu + b * Dd;
  float ss = 0.f, sw = 0.f;
  for (int d = lane; d < Dd; d += 32) {
    float x  = xp[d];
    float xc = x - mp[d];
    ss += xc * xc;
    sw += x * W[d];
  }
  #pragma unroll
  for (int m = 1; m < 32; m <<= 1) {
    ss += __shfl_xor(ss, m, 32);
    sw += __shfl_xor(sw, m, 32);
  }
  if (lane == 0) {
    float v = sqrtf(ss) + 1e-6f;
    var[row] = v;
    xw[row]  = sw;
    w1[row]  = 1.0f / v;
  }
}

// ---------------- Heavy pass: acc[slot,row] = sum_m |cov[row,m]| * w[m] -----------
// cov tiles computed on the fly with V_WMMA_F32_16X16X4_F32.
// Block = 128 threads (4 waves, 2x2). Each wave: 32x32 output = 2x2 WMMA tiles,
// 4 independent accumulator chains, each LDS fragment reused by 2 WMMAs.
// LDS double-buffered: one barrier per K-chunk, loads overlap compute.
__global__ __launch_bounds__(128)
void corr_pass_kernel(const float* __restrict__ X, const float* __restrict__ mu,
                      const float* __restrict__ w, float* __restrict__ acc) {
  __shared__ float lA[2][BT * LSTR];
  __shared__ float lB[2][BT * LSTR];

  const int b    = blockIdx.y;
  const int n0   = blockIdx.x * BT;
  const int tid  = threadIdx.x;
  const int wave = tid >> 5;
  const int lane = tid & 31;
  const int wn   = wave >> 1;        // 0..1 : 32-row band of the n-tile
  const int wm   = wave & 1;         // 0..1 : 32-col band of the m-tile
  const int lrow = lane & 15;        // fragment row (M for A, N for B)
  const int kh   = lane >> 4;        // K half: ks = 2*kh + {0,1}

  const float* Xb  = X  + (size_t)b * Nn * Dd;
  const float* mub = mu + (size_t)b * Dd;

  // cooperative mean-centered tile fill: 64x32 floats for A and B each
  auto fill = [&](int buf, int k0, int m0) {
    #pragma unroll
    for (int it = 0; it < 4; ++it) {
      int item = tid + it * 128;          // 0..511
      int row  = item >> 3;               // 0..63
      int col  = (item & 7) << 2;         // 0..28 step 4
      const float4 m4 = *(const float4*)(mub + k0 + col);
      const float4 xa = *(const float4*)(Xb + (size_t)(n0 + row) * Dd + k0 + col);
      const float4 xb = *(const float4*)(Xb + (size_t)(m0 + row) * Dd + k0 + col);
      float4 ra, rb;
      ra.x = xa.x - m4.x; ra.y = xa.y - m4.y; ra.z = xa.z - m4.z; ra.w = xa.w - m4.w;
      rb.x = xb.x - m4.x; rb.y = xb.y - m4.y; rb.z = xb.z - m4.z; rb.w = xb.w - m4.w;
      *(float4*)(&lA[buf][row * LSTR + col]) = ra;
      *(float4*)(&lB[buf][row * LSTR + col]) = rb;
    }
  };

  float rowpart[16];
  #pragma unroll
  for (int r = 0; r < 16; ++r) rowpart[r] = 0.f;

  for (int m0 = 0; m0 < Nn; m0 += BT) {
    v8f a00 = {}, a01 = {}, a10 = {}, a11 = {};

    fill(0, 0, m0);
    __syncthreads();

    for (int kc = 0; kc < NC; kc += 2) {
      // ---- chunk kc (buffer 0), prefetch kc+1 into buffer 1 ----
      fill(1, (kc + 1) * KC, m0);
      {
        const float* pa0 = &lA[0][(wn * 32 + lrow) * LSTR + 2 * kh];
        const float* pa1 = pa0 + 16 * LSTR;
        const float* pb0 = &lB[0][(wm * 32 + lrow) * LSTR + 2 * kh];
        const float* pb1 = pb0 + 16 * LSTR;
        #pragma unroll
        for (int kk = 0; kk < KC; kk += 4) {
          v2f fa0 = *(const v2f*)(pa0 + kk);
          v2f fa1 = *(const v2f*)(pa1 + kk);
          v2f fb0 = *(const v2f*)(pb0 + kk);
          v2f fb1 = *(const v2f*)(pb1 + kk);
          a00 = __builtin_amdgcn_wmma_f32_16x16x4_f32(false, fa0, false, fb0, (short)0, a00, false, false);
          a01 = __builtin_amdgcn_wmma_f32_16x16x4_f32(false, fa0, false, fb1, (short)0, a01, false, false);
          a10 = __builtin_amdgcn_wmma_f32_16x16x4_f32(false, fa1, false, fb0, (short)0, a10, false, false);
          a11 = __builtin_amdgcn_wmma_f32_16x16x4_f32(false, fa1, false, fb1, (short)0, a11, false, false);
        }
      }
      __syncthreads();

      // ---- chunk kc+1 (buffer 1), prefetch kc+2 into buffer 0 ----
      if (kc + 2 < NC) fill(0, (kc + 2) * KC, m0);
      {
        const float* pa0 = &lA[1][(wn * 32 + lrow) * LSTR + 2 * kh];
        const float* pa1 = pa0 + 16 * LSTR;
        const float* pb0 = &lB[1][(wm * 32 + lrow) * LSTR + 2 * kh];
        const float* pb1 = pb0 + 16 * LSTR;
        #pragma unroll
        for (int kk = 0; kk < KC; kk += 4) {
          v2f fa0 = *(const v2f*)(pa0 + kk);
          v2f fa1 = *(const v2f*)(pa1 + kk);
          v2f fb0 = *(const v2f*)(pb0 + kk);
          v2f fb1 = *(const v2f*)(pb1 + kk);
          a00 = __builtin_amdgcn_wmma_f32_16x16x4_f32(false, fa0, false, fb0, (short)0, a00, false, false);
          a01 = __builtin_amdgcn_wmma_f32_16x16x4_f32(false, fa0, false, fb1, (short)0, a01, false, false);
          a10 = __builtin_amdgcn_wmma_f32_16x16x4_f32(false, fa1, false, fb0, (short)0, a10, false, false);
          a11 = __builtin_amdgcn_wmma_f32_16x16x4_f32(false, fa1, false, fb1, (short)0, a11, false, false);
        }
      }
      __syncthreads();
    }

    // apply |.| and per-column weight; lane's values share column N = lrow
    const float wv0 = w[b * Nn + m0 + wm * 32 + lrow];        // tm = 0
    const float wv1 = w[b * Nn + m0 + wm * 32 + 16 + lrow];   // tm = 1
    #pragma unroll
    for (int r = 0; r < 8; ++r) {
      rowpart[r]     += fabsf(a00[r]) * wv0 + fabsf(a01[r]) * wv1;
      rowpart[8 + r] += fabsf(a10[r]) * wv0 + fabsf(a11[r]) * wv1;
    }
  }

  // reduce over the 16 lanes of each half (half kh holds rows M = 8*kh + r)
  #pragma unroll
  for (int r = 0; r < 16; ++r) {
    float v = rowpart[r];
    v += __shfl_xor(v, 1, 32);
    v += __shfl_xor(v, 2, 32);
    v += __shfl_xor(v, 4, 32);
    v += __shfl_xor(v, 8, 32);
    rowpart[r] = v;
  }
  if (lrow == 0) {
    #pragma unroll
    for (int tn = 0; tn < 2; ++tn) {
      #pragma unroll
      for (int r = 0; r < 8; ++r) {
        int row = n0 + wn * 32 + tn * 16 + 8 * kh + r;
        // private slot per wm -> plain store, deterministic (no atomics)
        acc[(size_t)wm * (Bb * Nn) + b * Nn + row] = rowpart[tn * 8 + r];
      }
    }
  }
}

// ---------------- finalize pass 1: dinv, w2 ----------------
__global__ void finalize1_kernel(const float* __restrict__ acc1, const float* __restrict__ var,
                                 const float* __restrict__ xw, float* __restrict__ dinv,
                                 float* __restrict__ w2) {
  int i = blockIdx.x * blockDim.x + threadIdx.x;
  if (i >= Bb * Nn) return;
  float v  = var[i];
  float R  = (acc1[i] + acc1[Bb * Nn + i]) / v;
  float di = 1.0f / sqrtf(R);
  dinv[i]  = di;
  w2[i]    = di * xw[i] / v;
}

// ---------------- finalize pass 2: S -> d_out[0 : B*N] ----------------
__global__ void finalize2_kernel(const float* __restrict__ acc2, const float* __restrict__ var,
                                 const float* __restrict__ dinv, float* __restrict__ S) {
  int i = blockIdx.x * blockDim.x + threadIdx.x;
  if (i >= Bb * Nn) return;
  S[i] = (acc2[i] + acc2[Bb * Nn + i]) * dinv[i] / var[i];
}

// ---------------- top-k (descending values, ascending index on ties) --------------
__global__ __launch_bounds__(256)
void topk_kernel(const float* __restrict__ S, int* __restrict__ topk) {
  __shared__ float sval[Nn];
  __shared__ float rv[256];
  __shared__ int   ri[256];
  const int b = blockIdx.x, tid = threadIdx.x;
  for (int i = tid; i < Nn; i += 256) sval[i] = S[b * Nn + i];
  __syncthreads();
  for (int k = 0; k < TOPK; ++k) {
    float best = -__builtin_inff(); int bi = Nn;
    for (int i = tid; i < Nn; i += 256) {
      float v = sval[i];
      if (v > best || (v == best && i < bi)) { best = v; bi = i; }
    }
    rv[tid] = best; ri[tid] = bi;
    __syncthreads();
    for (int s = 128; s > 0; s >>= 1) {
      if (tid < s) {
        float ov = rv[tid + s]; int oi = ri[tid + s];
        if (ov > rv[tid] || (ov == rv[tid] && oi < ri[tid])) { rv[tid] = ov; ri[tid] = oi; }
      }
      __syncthreads();
    }
    if (tid == 0) { topk[b * TOPK + k] = ri[0]; sval[ri[0]] = -__builtin_inff(); }
    __syncthreads();
  }
}

// ---------------- gather F_c rows ----------------
__global__ __launch_bounds__(256)
void gather_kernel(const float* __restrict__ X, const int* __restrict__ topk,
                   float* __restrict__ Fc) {
  int b = blockIdx.y, i = blockIdx.x;
  int idx = topk[b * TOPK + i];
  const float* src = X  + ((size_t)b * Nn + idx) * Dd;
  float*       dst = Fc + ((size_t)b * TOPK + i) * Dd;
  for (int d = threadIdx.x; d < Dd; d += 256) dst[d] = src[d];
}

// ---------------- F_c mean and F_nc mean (via total = N*mu) ----------------
__global__ void means_kernel(const float* __restrict__ X, const int* __restrict__ topk,
                             const float* __restrict__ mu, float* __restrict__ fc_mean,
                             float* __restrict__ fnc_mean) {
  int i = blockIdx.x * blockDim.x + threadIdx.x;
  if (i >= Bb * Dd) return;
  int b = i / Dd, d = i - b * Dd;
  float s = 0.f;
  for (int j = 0; j < TOPK; ++j)
    s += X[((size_t)b * Nn + topk[b * TOPK + j]) * Dd + d];
  fc_mean[i]  = s * (1.0f / TOPK);
  fnc_mean[i] = ((float)Nn * mu[i] - s) * (1.0f / (Nn - TOPK));
}

extern "C" void kernel_launch(void* const* d_in, const int* in_sizes, int n_in,
                              void* d_out, int out_size, void* d_ws, size_t ws_size,
                              hipStream_t stream) {
  (void)in_sizes; (void)n_in; (void)out_size; (void)ws_size;
  const float* X = (const float*)d_in[0];   // (8, 4096, 768)
  const float* W = (const float*)d_in[1];   // (1, 768)

  float* out = (float*)d_out;
  float* S_out    = out;                                  // B*N      = 32768
  float* Fc_out   = out + Bb * Nn;                        // B*K*D    = 307200
  float* FcM_out  = Fc_out + Bb * TOPK * Dd;              // B*D      = 6144
  float* FncM_out = FcM_out + Bb * Dd;                    // B*D      = 6144

  float* ws = (float*)d_ws;
  float* mu   = ws;                  // 6144
  float* var  = mu   + Bb * Dd;      // 32768
  float* xw   = var  + Bb * Nn;      // 32768
  float* w1   = xw   + Bb * Nn;      // 32768
  float* acc1 = w1   + Bb * Nn;      // 2*32768
  float* dinv = acc1 + 2 * Bb * Nn;  // 32768
  float* w2   = dinv + Bb * Nn;      // 32768
  float* acc2 = w2   + Bb * Nn;      // 2*32768
  int*   tki  = (int*)(acc2 + 2 * Bb * Nn);  // 400 ints

  mean_kernel<<<(Bb * Dd + 255) / 256, 256, 0, stream>>>(X, mu);
  row_stats_kernel<<<(Bb * Nn) / 8, 256, 0, stream>>>(X, W, mu, var, xw, w1);

  dim3 cgrid(Nn / BT, Bb);
  corr_pass_kernel<<<cgrid, 128, 0, stream>>>(X, mu, w1, acc1);
  finalize1_kernel<<<(Bb * Nn + 255) / 256, 256, 0, stream>>>(acc1, var, xw, dinv, w2);
  corr_pass_kernel<<<cgrid, 128, 0, stream>>>(X, mu, w2, acc2);
  finalize2_kernel<<<(Bb * Nn + 255) / 256, 256, 0, stream>>>(acc2, var, dinv, S_out);

  topk_kernel<<<Bb, 256, 0, stream>>>(S_out, tki);
  dim3 ggrid(TOPK, Bb);
  gather_kernel<<<ggrid, 256, 0, stream>>>(X, tki, Fc_out);
  means_kernel<<<(Bb * Dd + 255) / 256, 256, 0, stream>>>(X, tki, mu, FcM_out, FncM_out);
}